// DotAttention_29944511987940
// MI455X (gfx1250) — compile-verified
//
#include <hip/hip_runtime.h>
#include <math.h>

#define BATCH 64
#define LSEQ  4096
#define DDIM  1024
#define SPLITS 8
#define ROWS_PER_BLOCK (LSEQ / SPLITS)      // 512 rows per block
#define WAVES 8
#define BLOCK (WAVES * 32)                  // 256 threads
#define GROUPS_PER_WAVE (ROWS_PER_BLOCK / (16 * WAVES)) // 4 groups of 16 rows
#define MASK_VALUE (-50000.0f)
#define WS_STRIDE 1040                      // floats per partial: [0]=m,[1]=s, acc @ +8

typedef float v2f __attribute__((ext_vector_type(2)));
typedef float v4f __attribute__((ext_vector_type(4)));
typedef float v8f __attribute__((ext_vector_type(8)));

// ---------------- Phase A: fused logits (WMMA) + online softmax accumulate ----
__global__ __launch_bounds__(BLOCK) void attn_phaseA(
    const float* __restrict__ h, const int* __restrict__ mask,
    const float* __restrict__ s, float* __restrict__ wtmp,
    float* __restrict__ ws)
{
    __shared__ float s_lds[DDIM];
    __shared__ float wm[WAVES];
    __shared__ float wsum[WAVES];
    __shared__ v4f   buf[WAVES * (DDIM / 4)];   // 32 KB wave-partial acc

    const int tid   = threadIdx.x;
    const int wave  = tid >> 5;
    const int lane  = tid & 31;
    const int b     = blockIdx.x / SPLITS;
    const int split = blockIdx.x % SPLITS;
    const int rowStart0 = split * ROWS_PER_BLOCK;

    // decoder state -> LDS (coalesced, 256 * float4 = 1024 floats)
    ((v4f*)s_lds)[tid] = ((const v4f*)(s + (size_t)b * DDIM))[tid];
    __syncthreads();

    const int koff = (lane >> 4) * 2;   // K offset within 4-chunk per WMMA A/B layout
    const int mrow = lane & 15;         // M row owned by this lane

    float m   = -INFINITY;
    float sum = 0.0f;
    v4f   acc[8];
#pragma unroll
    for (int cc = 0; cc < 8; ++cc) acc[cc] = (v4f){0.f, 0.f, 0.f, 0.f};

    for (int t = 0; t < GROUPS_PER_WAVE; ++t) {
        const int rowBase = rowStart0 + (wave + WAVES * t) * 16;
        const size_t rowBaseIdx = (size_t)b * LSEQ + rowBase;

        // coalesced mask fetch for the 16 rows, issued early so the WMMA
        // k-loop hides its latency; broadcast per-row later via shfl
        const int mk = mask[rowBaseIdx + mrow];

        const float* aPtr = h + (rowBaseIdx + mrow) * DDIM + koff;
        const float* bPtr = s_lds + koff;

        // ---- logits for 16 rows via V_WMMA_F32_16X16X4_F32, K-loop over D ----
        v8f c8 = {0.f, 0.f, 0.f, 0.f, 0.f, 0.f, 0.f, 0.f};
#pragma unroll 8
        for (int k0 = 0; k0 < DDIM; k0 += 4) {
            v2f a  = *(const v2f*)(aPtr + k0);   // A: lane m holds row m, K={koff,koff+1}
            v2f bb = *(const v2f*)(bPtr + k0);   // B: broadcast s chunk (all N columns equal)
            c8 = __builtin_amdgcn_wmma_f32_16x16x4_f32(
                false, a, false, bb, (short)0, c8, false, false);
        }

        // broadcast the 16 logits to every lane (C layout: reg j = rows j / j+8)
        float lg[16];
#pragma unroll
        for (int j = 0; j < 8; ++j) {
            lg[j]     = __shfl(c8[j], 0, 32);
            lg[j + 8] = __shfl(c8[j], 16, 32);
        }

        // masked logits (wave-uniform values), mask broadcast from the lane fetch
        float ml[16];
#pragma unroll
        for (int r = 0; r < 16; ++r) {
            const int mkr = __shfl(mk, r, 32);
            ml[r] = mkr ? lg[r] : MASK_VALUE;
        }

        // stash masked logits for phase B: 4 packed b128 stores from lane 0
        if (lane == 0) {
            v4f* wp = (v4f*)(wtmp + rowBaseIdx);
            wp[0] = (v4f){ml[0],  ml[1],  ml[2],  ml[3]};
            wp[1] = (v4f){ml[4],  ml[5],  ml[6],  ml[7]};
            wp[2] = (v4f){ml[8],  ml[9],  ml[10], ml[11]};
            wp[3] = (v4f){ml[12], ml[13], ml[14], ml[15]};
        }

        // ---- per-row online-softmax accumulate (tile is L2-hot; last use -> NT) ----
#pragma unroll
        for (int r = 0; r < 16; ++r) {
            if (ml[r] > m) {
                const float sc = expf(m - ml[r]);
                sum *= sc;
#pragma unroll
                for (int cc = 0; cc < 8; ++cc) acc[cc] *= sc;
                m = ml[r];
            }
            const float e = expf(ml[r] - m);
            sum += e;
            const v4f* hp = (const v4f*)(h + (rowBaseIdx + r) * DDIM);
#pragma unroll
            for (int cc = 0; cc < 8; ++cc) {      // coalesced b128, dims lane*4+128*cc
                const v4f hv = __builtin_nontemporal_load(hp + lane + 32 * cc);
                acc[cc] += hv * e;
            }
        }
    }

    // ---- combine 8 wave partials -> one block partial (deterministic tree) ----
    if (lane == 0) { wm[wave] = m; wsum[wave] = sum; }
    __syncthreads();
    float M = -INFINITY;
#pragma unroll
    for (int j = 0; j < WAVES; ++j) M = fmaxf(M, wm[j]);
    const float f = expf(m - M);
#pragma unroll
    for (int cc = 0; cc < 8; ++cc)
        buf[wave * (DDIM / 4) + lane + 32 * cc] = acc[cc] * f;
    __syncthreads();

    float* base = ws + (size_t)(b * SPLITS + split) * WS_STRIDE;
    v4f o = (v4f){0.f, 0.f, 0.f, 0.f};
#pragma unroll
    for (int w = 0; w < WAVES; ++w)
        o += buf[w * (DDIM / 4) + tid];
    ((v4f*)(base + 8))[tid] = o;
    if (tid == 0) {
        float S = 0.0f;
        for (int j = 0; j < WAVES; ++j) S += wsum[j] * expf(wm[j] - M);
        base[0] = M;
        base[1] = S;
    }
}

// ---------------- Phase B: merge splits, write out + final weights -----------
__global__ __launch_bounds__(256) void attn_phaseB(
    const float* __restrict__ ws, float* __restrict__ out,
    float* __restrict__ wts)
{
    const int b   = blockIdx.x;
    const int tid = threadIdx.x;

    float mj[SPLITS], sj[SPLITS], fj[SPLITS];
    float M = -INFINITY;
#pragma unroll
    for (int j = 0; j < SPLITS; ++j) {
        const float* base = ws + (size_t)(b * SPLITS + j) * WS_STRIDE;
        mj[j] = base[0];
        sj[j] = base[1];
        M = fmaxf(M, mj[j]);
    }
    float S = 0.0f;
#pragma unroll
    for (int j = 0; j < SPLITS; ++j) { fj[j] = expf(mj[j] - M); S += sj[j] * fj[j]; }
    const float inv = 1.0f / S;

    // out[b, :] : 256 threads x float4 = 1024 dims
    v4f o = (v4f){0.f, 0.f, 0.f, 0.f};
#pragma unroll
    for (int j = 0; j < SPLITS; ++j) {
        const v4f v = ((const v4f*)(ws + (size_t)(b * SPLITS + j) * WS_STRIDE + 8))[tid];
        o += v * fj[j];
    }
    ((v4f*)(out + (size_t)b * DDIM))[tid] = o * inv;

    // weights: masked logits -> softmax (exp(MASK - M) underflows to exact 0)
    float* wrow = wts + (size_t)b * LSEQ;
    for (int l = tid; l < LSEQ; l += 256)
        wrow[l] = expf(wrow[l] - M) * inv;
}

extern "C" void kernel_launch(void* const* d_in, const int* in_sizes, int n_in,
                              void* d_out, int out_size, void* d_ws, size_t ws_size,
                              hipStream_t stream) {
    (void)in_sizes; (void)n_in; (void)out_size; (void)ws_size;
    const float* h    = (const float*)d_in[0];   // encoder_outputs [B,L,D]
    const int*   mask = (const int*)  d_in[1];   // mask [B,L]
    const float* s    = (const float*)d_in[2];   // decoder_state [B,D]
    float* out = (float*)d_out;                        // [B,D]
    float* wts = (float*)d_out + (size_t)BATCH * DDIM; // [B,L]
    float* ws  = (float*)d_ws;                         // SPLITS partials per batch

    attn_phaseA<<<BATCH * SPLITS, BLOCK, 0, stream>>>(h, mask, s, wts, ws);
    attn_phaseB<<<BATCH, 256, 0, stream>>>(ws, out, wts);
}